// MultiChannelGCNDecoder_48215302865297
// MI455X (gfx1250) — compile-verified
//
#include <hip/hip_runtime.h>

#define BATCH 16384
#define RCH 5
#define NN 68
#define KADJ 32
#define NL 3
#define LAT 68
#define NP 80            // padded node/latent dim (5 N-tiles of 16, 20 k-steps of 4)
#define XSTR 84          // x-tile row stride: 20*l mod 64 is a permutation -> no bank conflicts
#define TB 64            // batch rows per block
#define NWAVE 4
#define THREADS (NWAVE * 32)
#define NT 5             // N tiles of 16
#define NSTAGE 4         // fc + 3 gcn layers
#define TOTSTAGE (RCH * NSTAGE)
#define STAGE_F (NP * NP + NP)          // 6480 floats per stage block (W then bias)
#define STAGE_CHUNKS (STAGE_F / 4)      // 1620 b128 chunks (25920 B, 16B-divisible)

typedef float v2f __attribute__((ext_vector_type(2)));
typedef float v8f __attribute__((ext_vector_type(8)));

// ---------------------------------------------------------------------------
// Prep: build 20 dense, zero-padded stage blocks [W(80x80) | bias(80)].
// stage 0: fc_w[r];  stage 1..3: Wd[m,n] = sum_k [adj[n,k]==m] * gcn_w[r,l,n,k]
// ---------------------------------------------------------------------------
__global__ void gcn_prep(const float* __restrict__ fc_w,
                         const float* __restrict__ fc_b,
                         const int* __restrict__ adj,
                         const float* __restrict__ gcn_w,
                         const float* __restrict__ gcn_b,
                         float* __restrict__ WS) {
  const int mat = blockIdx.x;           // [0, TOTSTAGE)
  const int r = mat / NSTAGE;
  const int s = mat % NSTAGE;
  float* w = WS + mat * STAGE_F;
  float* bias = w + NP * NP;
  const int t = threadIdx.x;

  if (s == 0) {
    for (int idx = t; idx < NP * NP; idx += blockDim.x) {
      int m = idx / NP, n = idx % NP;
      w[idx] = (m < LAT && n < NN) ? fc_w[(r * LAT + m) * NN + n] : 0.f;
    }
    for (int n = t; n < NP; n += blockDim.x)
      bias[n] = (n < NN) ? fc_b[r * NN + n] : 0.f;
  } else {
    const int l = s - 1;
    for (int idx = t; idx < NP * NP; idx += blockDim.x) w[idx] = 0.f;
    __syncthreads();
    for (int idx = t; idx < NN * KADJ; idx += blockDim.x) {
      int n = idx / KADJ, k = idx % KADJ;
      int m = adj[n * KADJ + k];
      atomicAdd(&w[m * NP + n], gcn_w[((r * NL + l) * NN + n) * KADJ + k]);
    }
    for (int n = t; n < NP; n += blockDim.x)
      bias[n] = (n < NN) ? gcn_b[(r * NL + l) * NN + n] : 0.f;
  }
}

// ---------------------------------------------------------------------------
// Main kernel
// ---------------------------------------------------------------------------
__global__ __launch_bounds__(THREADS)
void gcn_main(const float* __restrict__ z, const float* __restrict__ WS,
              float* __restrict__ out) {
  __shared__ float zbuf[TB * XSTR];            // 21 KB
  __shared__ float xbuf[TB * XSTR];            // 21 KB
  __shared__ float wstage[STAGE_F];            // 25.9 KB  (W | bias)
  __shared__ float xs[RCH][TB * NN];           // 87 KB

  const int t = threadIdx.x;
  const int lane = t & 31;
  const int wave = t >> 5;
  const int b0 = blockIdx.x * TB;

  const int lhalf = lane >> 4;                 // 0: lanes 0-15, 1: lanes 16-31
  const int lmod = lane & 15;
  const int wrow = wave * 16;

  const unsigned wstage_lds = (unsigned)(uintptr_t)(&wstage[0]);

  // --- async prefetch of stage-0 weights (CDNA5 async-to-LDS, ASYNCcnt) ---
  {
    const float* src = WS;                     // uniform -> SGPR pair
    for (int c = t; c < STAGE_CHUNKS; c += THREADS) {
      unsigned voff = (unsigned)c * 16u;
      asm volatile("global_load_async_to_lds_b128 %0, %1, %2"
                   :: "v"(wstage_lds + voff), "v"(voff), "s"(src)
                   : "memory");
    }
  }

  // Stage the z tile (zero-padded columns), read-once -> nontemporal
  for (int idx = t; idx < TB * XSTR; idx += THREADS) {
    int row = idx / XSTR, c = idx % XSTR;
    zbuf[idx] = (c < LAT) ? __builtin_nontemporal_load(&z[(b0 + row) * LAT + c]) : 0.f;
  }

  asm volatile("s_wait_asynccnt 0x0" ::: "memory");
  __syncthreads();                             // wstage + zbuf ready for all waves

  for (int st = 0; st < TOTSTAGE; ++st) {
    const int r = st / NSTAGE;
    const int s = st % NSTAGE;
    const float* asrc = (s == 0) ? zbuf : xbuf;

    // Bias into registers now; these DS loads retire (in-order with the
    // k-loop's DS traffic) before the post-compute barrier, i.e. before the
    // next-stage prefetch can overwrite wstage.
    float biasreg[NT];
#pragma unroll
    for (int nt = 0; nt < NT; ++nt) biasreg[nt] = wstage[NP * NP + nt * 16 + lmod];

    v8f acc[NT] = {};
    for (int k0 = 0; k0 < NP; k0 += 4) {
      // A fragment (16x4 f32): lane holds M=lmod, K = k0 + 2*lhalf + {0,1}
      const float* ap = asrc + (wrow + lmod) * XSTR + k0 + 2 * lhalf;
      v2f a;
      a.x = ap[0];
      a.y = ap[1];
#pragma unroll
      for (int nt = 0; nt < NT; ++nt) {
        // B fragment (4x16 f32): vgpr v holds K = k0 + 2*lhalf + v, N = lmod
        v2f bf;
        bf.x = wstage[(k0 + 2 * lhalf + 0) * NP + nt * 16 + lmod];
        bf.y = wstage[(k0 + 2 * lhalf + 1) * NP + nt * 16 + lmod];
        acc[nt] = __builtin_amdgcn_wmma_f32_16x16x4_f32(
            false, a, false, bf, (short)0, acc[nt], false, false);
      }
    }

    __syncthreads();   // every wave done reading wstage and asrc

    // Prefetch next stage's W+bias; overlaps with the epilogue below.
    if (st + 1 < TOTSTAGE) {
      const float* src = WS + (size_t)(st + 1) * STAGE_F;
      for (int c = t; c < STAGE_CHUNKS; c += THREADS) {
        unsigned voff = (unsigned)c * 16u;
        asm volatile("global_load_async_to_lds_b128 %0, %1, %2"
                     :: "v"(wstage_lds + voff), "v"(voff), "s"(src)
                     : "memory");
      }
    }

    // Epilogue: bias (+ sigmoid for GCN layers), write to xbuf or xs[r].
    // C/D layout: vgpr v -> M = v + 8*lhalf, N = nt*16 + lmod
#pragma unroll
    for (int nt = 0; nt < NT; ++nt) {
      const int n = nt * 16 + lmod;
#pragma unroll
      for (int v = 0; v < 8; ++v) {
        const int row = wrow + v + 8 * lhalf;
        const float val = acc[nt][v] + biasreg[nt];
        if (s == 0) {
          xbuf[row * XSTR + n] = val;                 // FC has no activation
        } else {
          const float sg = 1.f / (1.f + __expf(-val));
          if (s < NSTAGE - 1) {
            xbuf[row * XSTR + n] = sg;
          } else if (n < NN) {
            xs[r][row * NN + n] = sg;                 // final layer -> keep
          }
        }
      }
    }

    asm volatile("s_wait_asynccnt 0x0" ::: "memory");
    __syncthreads();   // next W ready; xbuf writeback visible
  }

  // Output: out[b,i,j] = sum_r xs[r][b,i] * xs[r][b,j]. The sum of outer
  // products is symmetric, so the reference's 0.5*(S+S^T) is an identity.
  // Lane-contiguous j -> coalesced stores; nontemporal (write-once, 303 MB).
  for (int row = wave; row < TB; row += NWAVE) {
    float xj[RCH][3];
#pragma unroll
    for (int c = 0; c < 3; ++c) {
      const int j = lane + 32 * c;
#pragma unroll
      for (int rr = 0; rr < RCH; ++rr)
        xj[rr][c] = (j < NN) ? xs[rr][row * NN + j] : 0.f;
    }
    float* orow = out + (size_t)(b0 + row) * (NN * NN);
    for (int i = 0; i < NN; ++i) {
      float xi[RCH];
#pragma unroll
      for (int rr = 0; rr < RCH; ++rr) xi[rr] = xs[rr][row * NN + i];  // broadcast
#pragma unroll
      for (int c = 0; c < 3; ++c) {
        const int j = lane + 32 * c;
        if (j < NN) {
          float acc = 0.f;
#pragma unroll
          for (int rr = 0; rr < RCH; ++rr) acc = fmaf(xi[rr], xj[rr][c], acc);
          __builtin_nontemporal_store(acc, &orow[i * NN + j]);
        }
      }
    }
  }
}

// ---------------------------------------------------------------------------
extern "C" void kernel_launch(void* const* d_in, const int* in_sizes, int n_in,
                              void* d_out, int out_size, void* d_ws, size_t ws_size,
                              hipStream_t stream) {
  const float* z     = (const float*)d_in[0];
  const int*   adj   = (const int*)d_in[1];
  const float* fc_w  = (const float*)d_in[2];
  const float* fc_b  = (const float*)d_in[3];
  const float* gcn_w = (const float*)d_in[4];
  const float* gcn_b = (const float*)d_in[5];

  float* WS = (float*)d_ws;   // 20 stages * 6480 f32 = 518400 B

  gcn_prep<<<TOTSTAGE, 256, 0, stream>>>(fc_w, fc_b, adj, gcn_w, gcn_b, WS);
  gcn_main<<<BATCH / TB, THREADS, 0, stream>>>(z, WS, (float*)d_out);
}